// GCNClassifier_65154653880488
// MI455X (gfx1250) — compile-verified
//
#include <hip/hip_runtime.h>
#include <hip/hip_bf16.h>

#define N_NODES  100000
#define N_EDGES  1600000
#define N_GRAPHS 64
#define D        128
#define EPS      1e-5f

typedef float v2f __attribute__((ext_vector_type(2)));
typedef float v8f __attribute__((ext_vector_type(8)));
typedef int   v4i __attribute__((vector_size(16)));   // async-to-LDS builtin operand type

#if defined(__AMDGCN__) && __has_builtin(__builtin_amdgcn_global_load_async_to_lds_b128) && \
    __has_builtin(__builtin_amdgcn_s_wait_asynccnt)
#define USE_ASYNC_LDS 1
#else
#define USE_ASYNC_LDS 0
#endif

// ---------------- utility ----------------
__global__ void zero_f(float* __restrict__ p, long long n) {
    long long i = (long long)blockIdx.x * blockDim.x + threadIdx.x;
    long long stride = (long long)gridDim.x * blockDim.x;
    for (; i < n; i += stride) p[i] = 0.0f;
}

// ---------------- degrees / norms ----------------
__global__ __launch_bounds__(256) void degree_kernel(const int* __restrict__ src,
                                                     const int* __restrict__ dst,
                                                     float* __restrict__ deg_out,
                                                     float* __restrict__ deg_in) {
    int e = blockIdx.x * 256 + threadIdx.x;
    if (e >= N_EDGES) return;
    atomicAdd(&deg_out[src[e]], 1.0f);
    atomicAdd(&deg_in[dst[e]], 1.0f);
}

__global__ __launch_bounds__(256) void norm_kernel(float* __restrict__ buf, int n) {
    int i = blockIdx.x * 256 + threadIdx.x;
    if (i >= n) return;
    buf[i] = rsqrtf(fmaxf(buf[i], 1.0f));
}

// ---------------- embedding gather: h[n, :] = embed[tokens[n], :] ----------------
__global__ __launch_bounds__(256) void gather_kernel(const int* __restrict__ tokens,
                                                     const float* __restrict__ embed,
                                                     float* __restrict__ h) {
    long long t = (long long)blockIdx.x * 256 + threadIdx.x;
    int n = (int)(t >> 7);
    int c = (int)(t & 127);
    if (n >= N_NODES) return;
    h[(long long)n * D + c] = embed[(long long)tokens[n] * D + c];
}

// ---------------- edge scatter: agg[dst] += h[src] * norm_src[src] ----------------
__global__ __launch_bounds__(256) void scatter_kernel(const int* __restrict__ src,
                                                      const int* __restrict__ dst,
                                                      const float* __restrict__ h,
                                                      const float* __restrict__ nsrc,
                                                      float* __restrict__ agg) {
    long long t = (long long)blockIdx.x * 256 + threadIdx.x;
    int e = (int)(t >> 7);
    int c = (int)(t & 127);
    if (e >= N_EDGES) return;
    int s = src[e];
    int d = dst[e];
    atomicAdd(&agg[(long long)d * D + c], h[(long long)s * D + c] * nsrc[s]);
}

// ---------------- WMMA GEMM: y = diag(norm_dst) * (agg @ W) + b ; fused BN stats
// Block = 256 threads = 8 wave32s. Block owns 16 rows; wave w owns columns
// [16w, 16w+16). X tile staged in LDS via async global->LDS DMA when available
// (stride 132 floats -> conflict-free A reads). Inner loop: 32x
// v_wmma_f32_16x16x4_f32 over K=128, no VALU in the loop; the per-row
// norm_dst scale commutes with the GEMM and is applied to the accumulator
// in the epilogue (8 mults/lane/tile instead of 64 in-loop).
__global__ __launch_bounds__(256) void gemm_bn_kernel(const float* __restrict__ agg,
                                                      const float* __restrict__ ndst,
                                                      const float* __restrict__ W,
                                                      const float* __restrict__ bias,
                                                      float* __restrict__ y,
                                                      float* __restrict__ colsum,
                                                      float* __restrict__ colsumsq) {
    __shared__ float xs[16 * 132];
    const int tid  = threadIdx.x;
    const int row0 = blockIdx.x * 16;

#if USE_ASYNC_LDS
    // Async DMA stage: 16 rows x 128 cols, 2x b128 per thread, no VGPR round-trip.
    {
        int r  = tid >> 4;          // 0..15
        int c0 = (tid & 15) * 8;    // 8 floats per thread
        const float* gp = agg + (long long)(row0 + r) * D + c0;
        float* lp = &xs[r * 132 + c0];
        __builtin_amdgcn_global_load_async_to_lds_b128((v4i*)gp, (v4i*)lp, 0, 0);
        __builtin_amdgcn_global_load_async_to_lds_b128((v4i*)(gp + 4), (v4i*)(lp + 4), 0, 0);
    }
    __builtin_amdgcn_s_wait_asynccnt(0);
    __syncthreads();
#else
    // Fallback stage through VGPRs.
    {
        int r  = tid >> 4;
        int c0 = (tid & 15) * 8;
        const float* sp = agg + (long long)(row0 + r) * D + c0;
        #pragma unroll
        for (int i = 0; i < 8; i++) xs[r * 132 + c0 + i] = sp[i];
    }
    __syncthreads();
#endif

    const int lane = tid & 31;
    const int wid  = tid >> 5;      // 0..7 -> column tile
    const int n0   = wid * 16;
    const int hf   = lane >> 4;     // half-wave: 0 or 1
    const int mn   = lane & 15;     // row index (A) / col index (B,C)

    v8f acc = {};
    #pragma unroll
    for (int k0 = 0; k0 < D; k0 += 4) {
        int ka = k0 + hf * 2;       // K pair selected by half-wave
        v2f a, b;
        a.x = xs[mn * 132 + ka];
        a.y = xs[mn * 132 + ka + 1];
        const float* wp = W + (long long)ka * D + n0 + mn;
        b.x = wp[0];                // W[ka,   n0+mn]
        b.y = wp[D];                // W[ka+1, n0+mn]
        acc = __builtin_amdgcn_wmma_f32_16x16x4_f32(
            /*neg_a=*/false, a, /*neg_b=*/false, b,
            /*c_mod=*/(short)0, acc, /*reuse_a=*/false, /*reuse_b=*/false);
    }

    // C layout: VGPR r -> row = row0 + hf*8 + r, column = n0 + mn.
    // Apply the per-row norm_dst scale here (diag(s)*(X@W)), then bias + stats.
    const int col = n0 + mn;
    const float bv = bias[col];
    float s = 0.0f, s2 = 0.0f;
    #pragma unroll
    for (int r = 0; r < 8; r++) {
        int row = row0 + hf * 8 + r;
        float v = acc[r] * ndst[row] + bv;
        y[(long long)row * D + col] = v;
        s  += v;
        s2 += v * v;
    }
    atomicAdd(&colsum[col], s);
    atomicAdd(&colsumsq[col], s2);
}

// ---------------- BN (biased var) + ReLU ----------------
__global__ __launch_bounds__(256) void bn_relu_kernel(const float* __restrict__ y,
                                                      const float* __restrict__ colsum,
                                                      const float* __restrict__ colsumsq,
                                                      const float* __restrict__ gamma,
                                                      const float* __restrict__ beta,
                                                      float* __restrict__ h) {
    long long i = (long long)blockIdx.x * 256 + threadIdx.x;
    if (i >= (long long)N_NODES * D) return;
    int c = (int)(i & 127);
    const float invN = 1.0f / (float)N_NODES;
    float mu  = colsum[c] * invN;
    float var = colsumsq[c] * invN - mu * mu;
    float v = (y[i] - mu) * rsqrtf(var + EPS) * gamma[c] + beta[c];
    h[i] = fmaxf(v, 0.0f);
}

// ---------------- graph mean-pool accumulation ----------------
__global__ __launch_bounds__(256) void pool_kernel(const int* __restrict__ gid,
                                                   const float* __restrict__ h,
                                                   float* __restrict__ gsum,
                                                   float* __restrict__ gcnt) {
    long long t = (long long)blockIdx.x * 256 + threadIdx.x;
    int n = (int)(t >> 7);
    int c = (int)(t & 127);
    if (n >= N_NODES) return;
    int g = gid[n];
    atomicAdd(&gsum[(long long)g * D + c], h[(long long)n * D + c]);
    if (c == 0) atomicAdd(&gcnt[g], 1.0f);
}

// ---------------- FC head: [64,128] -> relu(128->64) -> (64->2) ----------------
__global__ __launch_bounds__(64) void head_kernel(const float* __restrict__ gsum,
                                                  const float* __restrict__ gcnt,
                                                  const float* __restrict__ fcW1,
                                                  const float* __restrict__ fcb1,
                                                  const float* __restrict__ fcW2,
                                                  const float* __restrict__ fcb2,
                                                  float* __restrict__ out) {
    __shared__ float hg[D];
    __shared__ float o1[64];
    int g = blockIdx.x;
    int j = threadIdx.x;       // 0..63
    float inv = 1.0f / fmaxf(gcnt[g], 1.0f);
    hg[j]      = gsum[g * D + j] * inv;
    hg[j + 64] = gsum[g * D + j + 64] * inv;
    __syncthreads();
    float acc = fcb1[j];
    #pragma unroll 4
    for (int k = 0; k < D; k++) acc += hg[k] * fcW1[k * 64 + j];
    o1[j] = fmaxf(acc, 0.0f);
    __syncthreads();
    if (j < 2) {
        float a = fcb2[j];
        for (int k = 0; k < 64; k++) a += o1[k] * fcW2[k * 2 + j];
        out[g * 2 + j] = a;
    }
}

// ---------------- host launch ----------------
extern "C" void kernel_launch(void* const* d_in, const int* in_sizes, int n_in,
                              void* d_out, int out_size, void* d_ws, size_t ws_size,
                              hipStream_t stream) {
    const int*   tokens = (const int*)d_in[0];
    const int*   src    = (const int*)d_in[1];
    const int*   dst    = (const int*)d_in[2];
    const int*   gids   = (const int*)d_in[3];
    const float* embed  = (const float*)d_in[4];
    const float* Wl[3]  = { (const float*)d_in[5],  (const float*)d_in[9],  (const float*)d_in[13] };
    const float* bl[3]  = { (const float*)d_in[6],  (const float*)d_in[10], (const float*)d_in[14] };
    const float* gl[3]  = { (const float*)d_in[7],  (const float*)d_in[11], (const float*)d_in[15] };
    const float* bel[3] = { (const float*)d_in[8],  (const float*)d_in[12], (const float*)d_in[16] };
    const float* fcW1   = (const float*)d_in[17];
    const float* fcb1   = (const float*)d_in[18];
    const float* fcW2   = (const float*)d_in[19];
    const float* fcb2   = (const float*)d_in[20];
    float* out = (float*)d_out;

    const long long ND = (long long)N_NODES * D;

    // workspace layout (floats): h | y | agg | colsum | colsumsq | nsrc | ndst | gsum | gcnt
    float* ws       = (float*)d_ws;
    float* h        = ws;
    float* y        = h + ND;
    float* agg      = y + ND;
    float* colsum   = agg + ND;          // contiguous with agg -> one zero pass
    float* colsumsq = colsum + D;
    float* nsrc     = colsumsq + D;
    float* ndst     = nsrc + N_NODES;    // contiguous with nsrc -> one zero pass
    float* gsum     = ndst + N_NODES;
    float* gcnt     = gsum + (long long)N_GRAPHS * D;  // contiguous with gsum

    // 1) degrees -> norms
    zero_f<<<1024, 256, 0, stream>>>(nsrc, 2LL * N_NODES);
    degree_kernel<<<(N_EDGES + 255) / 256, 256, 0, stream>>>(src, dst, nsrc, ndst);
    norm_kernel<<<(2 * N_NODES + 255) / 256, 256, 0, stream>>>(nsrc, 2 * N_NODES);

    // 2) embedding gather
    gather_kernel<<<(int)(ND / 256), 256, 0, stream>>>(tokens, embed, h);

    // 3) three GraphConv + BN + ReLU layers
    for (int l = 0; l < 3; l++) {
        zero_f<<<4096, 256, 0, stream>>>(agg, ND + 2 * D);                 // agg + BN stats
        scatter_kernel<<<(int)(((long long)N_EDGES * D) / 256), 256, 0, stream>>>(
            src, dst, h, nsrc, agg);
        gemm_bn_kernel<<<N_NODES / 16, 256, 0, stream>>>(
            agg, ndst, Wl[l], bl[l], y, colsum, colsumsq);
        bn_relu_kernel<<<(int)(ND / 256), 256, 0, stream>>>(
            y, colsum, colsumsq, gl[l], bel[l], h);
    }

    // 4) graph mean pooling + FC head
    zero_f<<<64, 256, 0, stream>>>(gsum, (long long)N_GRAPHS * D + N_GRAPHS);
    pool_kernel<<<(int)(ND / 256), 256, 0, stream>>>(gids, h, gsum, gcnt);
    head_kernel<<<N_GRAPHS, 64, 0, stream>>>(gsum, gcnt, fcW1, fcb1, fcW2, fcb2, out);
}